// TGCN_53249004536145
// MI455X (gfx1250) — compile-verified
//
#include <hip/hip_runtime.h>
#include <math.h>

// ---------------------------------------------------------------------------
// T-GCN cell (GCN z/r/h gates + GRU) for MI455X (gfx1250, wave32, WMMA).
// fp32 end-to-end using V_WMMA_F32_16X16X4_F32 to match the fp32 reference.
// 32-row M-tiles: each wave carries two accumulators and reuses every
// B (weight) fragment twice -> 2x WMMA:vmem ratio vs 16-row tiles.
// Workspace layout (floats): deg[N] | dinv[N] | S[N*128] | B1[N*128] |
// B2[N*128] | B3[N*128]  (~205.6 MB total for N=100000).
// ---------------------------------------------------------------------------

typedef __attribute__((ext_vector_type(2))) float v2f;
typedef __attribute__((ext_vector_type(8))) float v8f;

#define LDS_STRIDE 132   // 128 + 4 pad: rows land on distinct LDS banks
#define CDIM 128

__device__ __forceinline__ v8f wmma_f32(v2f a, v2f b, v8f c) {
    // D = A(16x4 f32) * B(4x16 f32) + C(16x16 f32)
    return __builtin_amdgcn_wmma_f32_16x16x4_f32(
        /*neg_a=*/false, a, /*neg_b=*/false, b,
        /*c_mod=*/(short)0, c, /*reuse_a=*/false, /*reuse_b=*/false);
}

__device__ __forceinline__ float sigmoidf_dev(float x) {
    return 1.0f / (1.0f + expf(-x));
}

__device__ __forceinline__ v2f lds_frag(const float* base, int row, int kk) {
    v2f a;
    a.x = base[row * LDS_STRIDE + kk];
    a.y = base[row * LDS_STRIDE + kk + 1];
    return a;
}

// --------------------------- degree / norm ---------------------------------

__global__ void k_deg_init(float* __restrict__ deg, int n) {
    int i = blockIdx.x * blockDim.x + threadIdx.x;
    if (i < n) deg[i] = 1.0f;   // self-loop weight (fill_value = 1)
}

__global__ void k_deg_acc(const int* __restrict__ dstv,
                          const float* __restrict__ ew,
                          float* __restrict__ deg, int e) {
    int i = blockIdx.x * blockDim.x + threadIdx.x;
    if (i < e) atomicAdd(&deg[dstv[i]], ew[i]);
}

__global__ void k_dinv(const float* __restrict__ deg,
                       float* __restrict__ dinv, int n) {
    int i = blockIdx.x * blockDim.x + threadIdx.x;
    if (i < n) {
        float d = deg[i];
        dinv[i] = (d > 0.0f) ? rsqrtf(d) : 0.0f;
    }
}

// --------------------------- GEMM1: S = X @ W ------------------------------
// grid.x = node tiles of 32 rows; block = 256 (8 waves); wave w -> cols 16w..
// Each wave computes two 16x16 tiles (row halves) sharing B fragments.
__global__ __launch_bounds__(256)
void k_gemm_xw(const float* __restrict__ X, const float* __restrict__ W,
               float* __restrict__ S, int n) {
    __shared__ float As[32 * LDS_STRIDE];
    const int row0 = blockIdx.x * 32;
    const int tid  = threadIdx.x;

    for (int idx = tid; idx < 32 * CDIM; idx += 256) {
        int r = idx >> 7, c = idx & 127;
        int row = row0 + r;
        As[r * LDS_STRIDE + c] = (row < n) ? X[(size_t)row * CDIM + c] : 0.0f;
    }
    __syncthreads();

    const int lane = tid & 31, w = tid >> 5;
    const int lo = lane & 15, hi = lane >> 4;
    const int col = w * 16 + lo;

    v8f acc0 = {0.f, 0.f, 0.f, 0.f, 0.f, 0.f, 0.f, 0.f};
    v8f acc1 = acc0;
    for (int k = 0; k < CDIM; k += 4) {
        const int kk = k + 2 * hi;
        v2f a0 = lds_frag(As, lo,      kk);
        v2f a1 = lds_frag(As, 16 + lo, kk);
        v2f b; b.x = W[(size_t)kk * CDIM + col];
               b.y = W[(size_t)(kk + 1) * CDIM + col];
        acc0 = wmma_f32(a0, b, acc0);
        acc1 = wmma_f32(a1, b, acc1);
    }
#pragma unroll
    for (int r = 0; r < 8; ++r) {
        int row = row0 + r + 8 * hi;
        if (row < n)      S[(size_t)row * CDIM + col]        = acc0[r];
        if (row + 16 < n) S[(size_t)(row + 16) * CDIM + col] = acc1[r];
    }
}

// ----------------- self-loop seed: B = S * dinv^2 (float4) -----------------

__global__ void k_selfloop(const float* __restrict__ S,
                           const float* __restrict__ dinv,
                           float* __restrict__ B, long total4) {
    long i = (long)blockIdx.x * blockDim.x + threadIdx.x;
    if (i >= total4) return;
    long node = i >> 5;                     // 32 float4 per 128-f row
    float d = dinv[node];
    float dd = d * d;
    float4 v = ((const float4*)S)[i];
    float4 o = {v.x * dd, v.y * dd, v.z * dd, v.w * dd};
    ((float4*)B)[i] = o;
}

// ------------- edge scatter: B[dst] += S[src] * norm (1 wave/edge) ---------

__global__ __launch_bounds__(256)
void k_edges(const int* __restrict__ srcv, const int* __restrict__ dstv,
             const float* __restrict__ ew, const float* __restrict__ dinv,
             const float* __restrict__ S, float* B, int e) {
    int wave = blockIdx.x * 8 + (threadIdx.x >> 5);
    if (wave >= e) return;
    int lane = threadIdx.x & 31;
    int s = srcv[wave];
    int d = dstv[wave];
    float norm = dinv[s] * ew[wave] * dinv[d];
    float4 v = ((const float4*)(S + (size_t)s * CDIM))[lane];
    float* drow = B + (size_t)d * CDIM + lane * 4;
    atomicAdd(drow + 0, v.x * norm);
    atomicAdd(drow + 1, v.y * norm);
    atomicAdd(drow + 2, v.z * norm);
    atomicAdd(drow + 3, v.w * norm);
}

// ---- Stage A: Z = sigmoid([Gz|H]@Lz + lz) -> Zout ; HR = H*sigmoid(R) -----
// Zout aliases accz, HRout aliases accr (reads staged to LDS before writes).
__global__ __launch_bounds__(256)
void k_stageA(const float* accz, const float* accr,
              const float* __restrict__ H,
              const float* __restrict__ bz, const float* __restrict__ br,
              const float* __restrict__ Lz, const float* __restrict__ lz,
              const float* __restrict__ Lr, const float* __restrict__ lr,
              float* Zout, float* HRout, int n) {
    __shared__ float Az[32 * LDS_STRIDE];
    __shared__ float Ar[32 * LDS_STRIDE];
    __shared__ float Ah[32 * LDS_STRIDE];
    const int row0 = blockIdx.x * 32;
    const int tid  = threadIdx.x;

    for (int idx = tid; idx < 32 * CDIM; idx += 256) {
        int r = idx >> 7, c = idx & 127;
        int row = row0 + r;
        if (row < n) {
            size_t g = (size_t)row * CDIM + c;
            Az[r * LDS_STRIDE + c] = accz[g] + bz[c];
            Ar[r * LDS_STRIDE + c] = accr[g] + br[c];
            Ah[r * LDS_STRIDE + c] = H[g];
        } else {
            Az[r * LDS_STRIDE + c] = 0.0f;
            Ar[r * LDS_STRIDE + c] = 0.0f;
            Ah[r * LDS_STRIDE + c] = 0.0f;
        }
    }
    __syncthreads();

    const int lane = tid & 31, w = tid >> 5;
    const int lo = lane & 15, hi = lane >> 4;
    const int col = w * 16 + lo;

    v8f cz0 = {0.f, 0.f, 0.f, 0.f, 0.f, 0.f, 0.f, 0.f};
    v8f cz1 = cz0, cr0 = cz0, cr1 = cz0;
    for (int k = 0; k < CDIM; k += 4) {
        const int kk = k + 2 * hi;
        v2f az0 = lds_frag(Az, lo, kk),      az1 = lds_frag(Az, 16 + lo, kk);
        v2f ar0 = lds_frag(Ar, lo, kk),      ar1 = lds_frag(Ar, 16 + lo, kk);
        v2f ah0 = lds_frag(Ah, lo, kk),      ah1 = lds_frag(Ah, 16 + lo, kk);
        // top half rows of L (gcn part), bottom half rows (H part)
        v2f bzt; bzt.x = Lz[(size_t)kk * CDIM + col];          bzt.y = Lz[(size_t)(kk + 1) * CDIM + col];
        v2f bzb; bzb.x = Lz[(size_t)(CDIM + kk) * CDIM + col]; bzb.y = Lz[(size_t)(CDIM + kk + 1) * CDIM + col];
        v2f brt; brt.x = Lr[(size_t)kk * CDIM + col];          brt.y = Lr[(size_t)(kk + 1) * CDIM + col];
        v2f brb; brb.x = Lr[(size_t)(CDIM + kk) * CDIM + col]; brb.y = Lr[(size_t)(CDIM + kk + 1) * CDIM + col];
        cz0 = wmma_f32(az0, bzt, cz0);  cz1 = wmma_f32(az1, bzt, cz1);
        cz0 = wmma_f32(ah0, bzb, cz0);  cz1 = wmma_f32(ah1, bzb, cz1);
        cr0 = wmma_f32(ar0, brt, cr0);  cr1 = wmma_f32(ar1, brt, cr1);
        cr0 = wmma_f32(ah0, brb, cr0);  cr1 = wmma_f32(ah1, brb, cr1);
    }

    float lzc = lz[col], lrc = lr[col];
#pragma unroll
    for (int r = 0; r < 8; ++r) {
        int rr0 = r + 8 * hi;
        int row = row0 + rr0;
        if (row < n) {
            size_t g = (size_t)row * CDIM + col;
            float z  = sigmoidf_dev(cz0[r] + lzc);
            float rg = sigmoidf_dev(cr0[r] + lrc);
            float h  = Ah[rr0 * LDS_STRIDE + col];
            Zout[g]  = z;
            HRout[g] = h * rg;
        }
        if (row + 16 < n) {
            size_t g = (size_t)(row + 16) * CDIM + col;
            float z  = sigmoidf_dev(cz1[r] + lzc);
            float rg = sigmoidf_dev(cr1[r] + lrc);
            float h  = Ah[(rr0 + 16) * LDS_STRIDE + col];
            Zout[g]  = z;
            HRout[g] = h * rg;
        }
    }
}

// ---- Stage B: Ht = tanh([Gh|HR]@Lh + lh); out = Z*H + (1-Z)*Ht ------------

__global__ __launch_bounds__(256)
void k_stageB(const float* __restrict__ acch, const float* __restrict__ HR,
              const float* __restrict__ Z, const float* __restrict__ H,
              const float* __restrict__ bh, const float* __restrict__ Lh,
              const float* __restrict__ lh, float* __restrict__ out, int n) {
    __shared__ float Ag[32 * LDS_STRIDE];
    __shared__ float Am[32 * LDS_STRIDE];
    const int row0 = blockIdx.x * 32;
    const int tid  = threadIdx.x;

    for (int idx = tid; idx < 32 * CDIM; idx += 256) {
        int r = idx >> 7, c = idx & 127;
        int row = row0 + r;
        if (row < n) {
            size_t g = (size_t)row * CDIM + c;
            Ag[r * LDS_STRIDE + c] = acch[g] + bh[c];
            Am[r * LDS_STRIDE + c] = HR[g];
        } else {
            Ag[r * LDS_STRIDE + c] = 0.0f;
            Am[r * LDS_STRIDE + c] = 0.0f;
        }
    }
    __syncthreads();

    const int lane = tid & 31, w = tid >> 5;
    const int lo = lane & 15, hi = lane >> 4;
    const int col = w * 16 + lo;

    v8f c0 = {0.f, 0.f, 0.f, 0.f, 0.f, 0.f, 0.f, 0.f};
    v8f c1 = c0;
    for (int k = 0; k < CDIM; k += 4) {
        const int kk = k + 2 * hi;
        v2f ag0 = lds_frag(Ag, lo, kk), ag1 = lds_frag(Ag, 16 + lo, kk);
        v2f am0 = lds_frag(Am, lo, kk), am1 = lds_frag(Am, 16 + lo, kk);
        v2f bt; bt.x = Lh[(size_t)kk * CDIM + col];          bt.y = Lh[(size_t)(kk + 1) * CDIM + col];
        v2f bb; bb.x = Lh[(size_t)(CDIM + kk) * CDIM + col]; bb.y = Lh[(size_t)(CDIM + kk + 1) * CDIM + col];
        c0 = wmma_f32(ag0, bt, c0);  c1 = wmma_f32(ag1, bt, c1);
        c0 = wmma_f32(am0, bb, c0);  c1 = wmma_f32(am1, bb, c1);
    }

    float lhc = lh[col];
#pragma unroll
    for (int r = 0; r < 8; ++r) {
        int row = row0 + r + 8 * hi;
        if (row < n) {
            size_t g  = (size_t)row * CDIM + col;
            float ht  = tanhf(c0[r] + lhc);
            float z   = Z[g];
            out[g] = z * H[g] + (1.0f - z) * ht;
        }
        if (row + 16 < n) {
            size_t g  = (size_t)(row + 16) * CDIM + col;
            float ht  = tanhf(c1[r] + lhc);
            float z   = Z[g];
            out[g] = z * H[g] + (1.0f - z) * ht;
        }
    }
}

// ---------------------------------------------------------------------------

extern "C" void kernel_launch(void* const* d_in, const int* in_sizes, int n_in,
                              void* d_out, int out_size, void* d_ws, size_t ws_size,
                              hipStream_t stream) {
    const float* X  = (const float*)d_in[0];
    const int*   ei = (const int*)d_in[1];
    const float* ew = (const float*)d_in[2];
    const float* H  = (const float*)d_in[3];
    const float* Wz = (const float*)d_in[4];
    const float* bz = (const float*)d_in[5];
    const float* Lz = (const float*)d_in[6];
    const float* lz = (const float*)d_in[7];
    const float* Wr = (const float*)d_in[8];
    const float* br = (const float*)d_in[9];
    const float* Lr = (const float*)d_in[10];
    const float* lr = (const float*)d_in[11];
    const float* Wh = (const float*)d_in[12];
    const float* bh = (const float*)d_in[13];
    const float* Lh = (const float*)d_in[14];
    const float* lh = (const float*)d_in[15];

    const int N = in_sizes[3] / CDIM;   // from H
    const int E = in_sizes[2];          // from edge_weight
    const int* srcv = ei;
    const int* dstv = ei + E;

    float* ws   = (float*)d_ws;
    size_t NN   = (size_t)N;
    float* deg  = ws;
    float* dinv = deg + NN;
    float* S    = dinv + NN;
    float* B1   = S  + NN * CDIM;   // acc_z -> later Z
    float* B2   = B1 + NN * CDIM;   // acc_r -> later H*R
    float* B3   = B2 + NN * CDIM;   // acc_h

    const int nTiles  = (N + 31) / 32;
    const int nB      = (N + 255) / 256;
    const int eB      = (E + 255) / 256;
    const int eWaveB  = (E + 7) / 8;          // 8 waves (1 edge each) per block
    const long total4 = (long)N * (CDIM / 4);
    const int sB      = (int)((total4 + 255) / 256);

    // symmetric normalization
    k_deg_init<<<nB, 256, 0, stream>>>(deg, N);
    k_deg_acc<<<eB, 256, 0, stream>>>(dstv, ew, deg, E);
    k_dinv<<<nB, 256, 0, stream>>>(deg, dinv, N);

    // gate z
    k_gemm_xw<<<nTiles, 256, 0, stream>>>(X, Wz, S, N);
    k_selfloop<<<sB, 256, 0, stream>>>(S, dinv, B1, total4);
    k_edges<<<eWaveB, 256, 0, stream>>>(srcv, dstv, ew, dinv, S, B1, E);

    // gate r
    k_gemm_xw<<<nTiles, 256, 0, stream>>>(X, Wr, S, N);
    k_selfloop<<<sB, 256, 0, stream>>>(S, dinv, B2, total4);
    k_edges<<<eWaveB, 256, 0, stream>>>(srcv, dstv, ew, dinv, S, B2, E);

    // Z and H*R (in-place into B1/B2)
    k_stageA<<<nTiles, 256, 0, stream>>>(B1, B2, H, bz, br, Lz, lz, Lr, lr,
                                         B1, B2, N);

    // gate h
    k_gemm_xw<<<nTiles, 256, 0, stream>>>(X, Wh, S, N);
    k_selfloop<<<sB, 256, 0, stream>>>(S, dinv, B3, total4);
    k_edges<<<eWaveB, 256, 0, stream>>>(srcv, dstv, ew, dinv, S, B3, E);

    // H_tilde + GRU blend
    k_stageB<<<nTiles, 256, 0, stream>>>(B3, B2, B1, H, bh, Lh, lh,
                                         (float*)d_out, N);
}